// GNNLSTM_14851996909757
// MI455X (gfx1250) — compile-verified
//
#include <hip/hip_runtime.h>
#include <hip/hip_bf16.h>

// ---------------------------------------------------------------------------
// GNN(2xGCN)->LSTM->FC on MI455X (gfx1250, wave32).
// Heavy math path: fused [10000,128]x[128,256] fp32 GEMM per time step using
// V_WMMA_F32_16X16X4_F32, with LSTM gate nonlinearities applied in-register.
// B operand pre-packed into WMMA lane order (one b64 load per WMMA).
// Graph aggregation via CSR gather (built once per launch), no float atomics.
// ---------------------------------------------------------------------------

#define NNODES 5000
#define NEDGES 160000
#define BB     2
#define TT     12
#define HH     64
#define RR     (BB * NNODES)   // 10000 LSTM rows

typedef __attribute__((ext_vector_type(2))) float v2f;
typedef __attribute__((ext_vector_type(8))) float v8f;

__device__ __forceinline__ float waveReduceAdd(float v) {
    #pragma unroll
    for (int off = 16; off > 0; off >>= 1)
        v += __shfl_xor(v, off, 32);
    return v;
}

__device__ __forceinline__ float sigf(float x) {
    return 1.0f / (1.0f + __expf(-x));
}

// ---------------- preprocessing kernels ----------------

__global__ void k_init_deg(int* deg) {
    int n = blockIdx.x * blockDim.x + threadIdx.x;
    if (n < NNODES) deg[n] = 1;   // self loop contributes 1
}

__global__ void k_count_deg(const int* ei, int* deg) {
    int e = blockIdx.x * blockDim.x + threadIdx.x;
    if (e < NEDGES) atomicAdd(&deg[ei[NEDGES + e]], 1);  // dst row of edge_index
}

__global__ void k_dinv(const int* deg, float* dinv) {
    int n = blockIdx.x * blockDim.x + threadIdx.x;
    if (n < NNODES) dinv[n] = rsqrtf((float)deg[n]);
}

// single-block exclusive scan of (deg-1) -> CSR row_start, cursor copy
__global__ void k_scan(const int* deg, int* row_start, int* cursor) {
    __shared__ int partial[1024];
    const int t = threadIdx.x;
    const int C = (NNODES + 1023) / 1024;
    int begin = t * C;
    int end = begin + C; if (end > NNODES) end = NNODES; if (begin > NNODES) begin = NNODES;
    int s = 0;
    for (int i = begin; i < end; ++i) s += deg[i] - 1;
    partial[t] = s;
    __syncthreads();
    for (int off = 1; off < 1024; off <<= 1) {
        int v = (t >= off) ? partial[t - off] : 0;
        __syncthreads();
        partial[t] += v;
        __syncthreads();
    }
    int prefix = (t == 0) ? 0 : partial[t - 1];
    for (int i = begin; i < end; ++i) {
        row_start[i] = prefix;
        cursor[i]    = prefix;
        prefix += deg[i] - 1;
    }
    if (t == 1023) row_start[NNODES] = partial[1023];  // == NEDGES
}

__global__ void k_fill_csr(const int* ei, const float* dinv, int* cursor,
                           int* csr_src, float* csr_w) {
    int e = blockIdx.x * blockDim.x + threadIdx.x;
    if (e >= NEDGES) return;
    int s = ei[e];
    int d = ei[NEDGES + e];
    int pos = atomicAdd(&cursor[d], 1);
    csr_src[pos] = s;
    csr_w[pos]   = dinv[s] * dinv[d];
}

// Wc[128,256]: top = W2 @ W_ih^T, bottom = W_hh^T ; bfull = b2@W_ih^T + b_ih + b_hh
__global__ void k_build_wc(const float* W2, const float* W_ih, const float* W_hh,
                           const float* b_ih, const float* b_hh, const float* b2,
                           float* Wc, float* bfull) {
    int j = blockIdx.x * blockDim.x + threadIdx.x;   // output gate column 0..255
    if (j >= 4 * HH) return;
    for (int k = 0; k < HH; ++k) {
        float s = 0.f;
        for (int m = 0; m < HH; ++m) s += W2[k * HH + m] * W_ih[j * HH + m];
        Wc[k * 256 + j] = s;
    }
    for (int k = 0; k < HH; ++k) Wc[(HH + k) * 256 + j] = W_hh[j * HH + k];
    float bs = b_ih[j] + b_hh[j];
    for (int k = 0; k < HH; ++k) bs += b2[k] * W_ih[j * HH + k];
    bfull[j] = bs;
}

// Repack Wc into WMMA-native B-fragment order:
// Wpk[((kblk*16 + ct)*32 + lane)] = { Wc[kblk*4 + lh*2][ct*16+lm],
//                                    Wc[kblk*4 + lh*2 + 1][ct*16+lm] }
// so each (k-step, tile) B operand is one coalesced 256B b64 load per wave.
__global__ void k_pack_wc(const float* __restrict__ Wc, float* __restrict__ Wpk) {
    int i = blockIdx.x * blockDim.x + threadIdx.x;   // 0 .. 32*16*32-1
    if (i >= 32 * 16 * 32) return;
    int lane = i & 31;
    int ct   = (i >> 5) & 15;
    int kblk = i >> 9;
    int lm = lane & 15, lh = lane >> 4;
    int r = kblk * 4 + lh * 2;
    int ccol = ct * 16 + lm;
    Wpk[2 * i + 0] = Wc[(size_t)r * 256 + ccol];
    Wpk[2 * i + 1] = Wc[(size_t)(r + 1) * 256 + ccol];
}

__global__ void k_zero(float* p, int n) {
    int i = blockIdx.x * blockDim.x + threadIdx.x;
    if (i < n) p[i] = 0.f;
}

// ---------------- per-step kernels ----------------

// GCN1: scalar aggregation (rank-1 trick) + outer product with W1 + bias + ReLU
__global__ void k_gcn1(const float* __restrict__ x, const int* __restrict__ row_start,
                       const int* __restrict__ csr_src, const float* __restrict__ csr_w,
                       const float* __restrict__ dinv, const float* __restrict__ W1,
                       const float* __restrict__ b1, float* __restrict__ h1, int t) {
    int wave = (blockIdx.x * blockDim.x + threadIdx.x) >> 5;
    int lane = threadIdx.x & 31;
    if (wave >= BB * NNODES) return;
    int b = wave / NNODES, n = wave - b * NNODES;
    const float* xb = x + (size_t)(b * TT + t) * NNODES;
    int rs = row_start[n], re = row_start[n + 1];
    float acc = 0.f;
    for (int e = rs + lane; e < re; e += 32)
        acc += csr_w[e] * xb[csr_src[e]];
    acc = waveReduceAdd(acc);
    float dn = dinv[n];
    float s = acc + dn * dn * xb[n];           // self-loop term
    float* out = h1 + (size_t)(b * NNODES + n) * HH;
    #pragma unroll
    for (int j = lane; j < HH; j += 32) {
        float v = s * W1[j] + b1[j];
        out[j] = v > 0.f ? v : 0.f;
    }
}

// GCN2 aggregation on 64-dim vectors (matmul folded into the LSTM GEMM later)
__global__ void k_gcn2agg(const float* __restrict__ h1, const int* __restrict__ row_start,
                          const int* __restrict__ csr_src, const float* __restrict__ csr_w,
                          const float* __restrict__ dinv, float* __restrict__ agg2) {
    int wave = (blockIdx.x * blockDim.x + threadIdx.x) >> 5;
    int lane = threadIdx.x & 31;
    if (wave >= BB * NNODES) return;
    int b = wave / NNODES, n = wave - b * NNODES;
    const float* base = h1 + (size_t)b * NNODES * HH;
    float dn = dinv[n];
    float w0 = dn * dn;
    float a0 = w0 * base[(size_t)n * HH + lane];
    float a1 = w0 * base[(size_t)n * HH + lane + 32];
    int rs = row_start[n], re = row_start[n + 1];
    for (int e = rs; e < re; ++e) {
        float w = csr_w[e];
        int   s = csr_src[e];
        a0 += w * base[(size_t)s * HH + lane];
        a1 += w * base[(size_t)s * HH + lane + 32];
    }
    float* out = agg2 + (size_t)(b * NNODES + n) * HH;
    out[lane]      = a0;
    out[lane + 32] = a1;
}

// Fused gates GEMM + LSTM cell update.
// One wave per workgroup (launch_bounds(32,1) -> large VGPR budget, no spills).
// Wave owns 16 rows x 256 gate cols: 16 v8f accumulators; tiles ct, ct+4,
// ct+8, ct+12 hold i,f,g,o for the SAME (row,col) in the same register slot,
// so the LSTM update is pure per-lane register math with zero shuffles.
__global__ void __launch_bounds__(32, 1)
k_lstm(const float* __restrict__ agg2, float* __restrict__ h,
       float* __restrict__ c, const float* __restrict__ Wpk,
       const float* __restrict__ bfull) {
    const int wave = blockIdx.x;
    const int lane = threadIdx.x;
    const int row0 = wave * 16;
    if (row0 >= RR) return;                 // wave-uniform: EXEC all-1 for WMMA
    const int lm = lane & 15;
    const int lh = lane >> 4;               // 0 or 1

    v8f acc[16];
    #pragma unroll
    for (int ct = 0; ct < 16; ++ct) {
        float bv = bfull[ct * 16 + lm];     // bias depends only on column
        acc[ct] = (v8f){bv, bv, bv, bv, bv, bv, bv, bv};
    }

    const int arow = row0 + lm;
    #pragma unroll 2
    for (int k = 0; k < 128; k += 4) {
        const float* Abase = (k < 64) ? agg2 : h;
        int kloc = (k < 64) ? k : (k - 64);
        // A 16x4 fragment: lanes 0-15 -> K=k..k+1, lanes 16-31 -> K=k+2..k+3
        v2f a = *(const v2f*)(Abase + (size_t)arow * HH + kloc + lh * 2);

        // batch all 16 B fragments first (coalesced b64 loads, clause-able),
        // then 16 back-to-back WMMAs
        const float* bp = Wpk + (size_t)(((k >> 2) * 16) * 32 + lane) * 2;
        v2f bb[16];
        #pragma unroll
        for (int ct = 0; ct < 16; ++ct)
            bb[ct] = *(const v2f*)(bp + (size_t)ct * 64);
        #pragma unroll
        for (int ct = 0; ct < 16; ++ct)
            acc[ct] = __builtin_amdgcn_wmma_f32_16x16x4_f32(
                false, a, false, bb[ct], (short)0, acc[ct], false, false);
    }

    #pragma unroll
    for (int ct = 0; ct < 4; ++ct) {
        #pragma unroll
        for (int v = 0; v < 8; ++v) {
            int row = row0 + v + lh * 8;
            int col = ct * 16 + lm;
            size_t idx = (size_t)row * HH + col;
            float iv = sigf(acc[ct][v]);
            float fv = sigf(acc[ct + 4][v]);
            float gv = tanhf(acc[ct + 8][v]);
            float ov = sigf(acc[ct + 12][v]);
            float cn = fv * c[idx] + iv * gv;
            float hn = ov * tanhf(cn);
            c[idx] = cn;
            h[idx] = hn;
        }
    }
}

__global__ void k_out(const float* __restrict__ h, const float* __restrict__ fc_w,
                      const float* __restrict__ fc_b, float* __restrict__ out) {
    int wave = (blockIdx.x * blockDim.x + threadIdx.x) >> 5;
    int lane = threadIdx.x & 31;
    if (wave >= RR) return;
    const float* hr = h + (size_t)wave * HH;
    float s = hr[lane] * fc_w[lane] + hr[lane + 32] * fc_w[lane + 32];
    s = waveReduceAdd(s);
    if (lane == 0) out[wave] = s + fc_b[0];
}

// ---------------- host side ----------------

static inline size_t align256(size_t x) { return (x + 255) & ~(size_t)255; }

extern "C" void kernel_launch(void* const* d_in, const int* in_sizes, int n_in,
                              void* d_out, int out_size, void* d_ws, size_t ws_size,
                              hipStream_t stream) {
    const float* x    = (const float*)d_in[0];
    const int*   ei   = (const int*)d_in[1];   // [2, E]: row0 = src, row1 = dst
    const float* W1   = (const float*)d_in[2];
    const float* b1   = (const float*)d_in[3];
    const float* W2   = (const float*)d_in[4];
    const float* b2   = (const float*)d_in[5];
    const float* W_ih = (const float*)d_in[6];
    const float* W_hh = (const float*)d_in[7];
    const float* b_ih = (const float*)d_in[8];
    const float* b_hh = (const float*)d_in[9];
    const float* fc_w = (const float*)d_in[10];
    const float* fc_b = (const float*)d_in[11];
    float* out = (float*)d_out;

    // workspace carve-out
    char* ws = (char*)d_ws;
    size_t off = 0;
    auto alloc = [&](size_t bytes) -> char* {
        char* p = ws + off;
        off = align256(off + bytes);
        return p;
    };
    int*   deg       = (int*)  alloc(NNODES * sizeof(int));
    float* dinv      = (float*)alloc(NNODES * sizeof(float));
    int*   row_start = (int*)  alloc((NNODES + 1) * sizeof(int));
    int*   cursor    = (int*)  alloc(NNODES * sizeof(int));
    int*   csr_src   = (int*)  alloc(NEDGES * sizeof(int));
    float* csr_w     = (float*)alloc(NEDGES * sizeof(float));
    float* Wc        = (float*)alloc(128 * 256 * sizeof(float));
    float* Wpk       = (float*)alloc(128 * 256 * sizeof(float));
    float* bfull     = (float*)alloc(256 * sizeof(float));
    float* h1        = (float*)alloc((size_t)RR * HH * sizeof(float));
    float* agg2      = (float*)alloc((size_t)RR * HH * sizeof(float));
    float* hbuf      = (float*)alloc((size_t)RR * HH * sizeof(float));
    float* cbuf      = (float*)alloc((size_t)RR * HH * sizeof(float));
    (void)ws_size; (void)n_in; (void)in_sizes; (void)out_size;

    const int TB = 256;
    // ---- preprocessing ----
    k_init_deg <<<(NNODES + TB - 1) / TB, TB, 0, stream>>>(deg);
    k_count_deg<<<(NEDGES + TB - 1) / TB, TB, 0, stream>>>(ei, deg);
    k_dinv     <<<(NNODES + TB - 1) / TB, TB, 0, stream>>>(deg, dinv);
    k_scan     <<<1, 1024, 0, stream>>>(deg, row_start, cursor);
    k_fill_csr <<<(NEDGES + TB - 1) / TB, TB, 0, stream>>>(ei, dinv, cursor, csr_src, csr_w);
    k_build_wc <<<1, 256, 0, stream>>>(W2, W_ih, W_hh, b_ih, b_hh, b2, Wc, bfull);
    k_pack_wc  <<<(32 * 16 * 32 + TB - 1) / TB, TB, 0, stream>>>(Wc, Wpk);
    {
        int n = RR * HH;
        k_zero<<<(n + TB - 1) / TB, TB, 0, stream>>>(hbuf, n);
        k_zero<<<(n + TB - 1) / TB, TB, 0, stream>>>(cbuf, n);
    }

    // ---- recurrent steps ----
    const int gcnWaves   = BB * NNODES;                       // 10000
    const int gcnBlocks  = (gcnWaves * 32 + TB - 1) / TB;     // 1250
    const int lstmBlocks = (RR + 15) / 16;                    // 625 single-wave WGs

    for (int t = 0; t < TT; ++t) {
        k_gcn1   <<<gcnBlocks, TB, 0, stream>>>(x, row_start, csr_src, csr_w,
                                                dinv, W1, b1, h1, t);
        k_gcn2agg<<<gcnBlocks, TB, 0, stream>>>(h1, row_start, csr_src, csr_w,
                                                dinv, agg2);
        k_lstm   <<<lstmBlocks, 32, 0, stream>>>(agg2, hbuf, cbuf, Wpk, bfull);
    }

    // ---- final projection ----
    k_out<<<(RR * 32 + TB - 1) / TB, TB, 0, stream>>>(hbuf, fc_w, fc_b, out);
}